// Model_69329362092344
// MI455X (gfx1250) — compile-verified
//
#include <hip/hip_runtime.h>

// ---------------------------------------------------------------------------
// MI455X (gfx1250) implementation.
//
// Roofline: ~95 MFLOP total, ~70 MB HBM traffic (g:25MB, f:40MB write,
// x:3.3MB); the 164MB f-gather stays inside the 192MB L2. Bandwidth bound,
// f32 end-to-end. WMMA f32 16x16x4 drives the 16384 gathered 50x50 matvecs.
// ---------------------------------------------------------------------------

typedef __attribute__((ext_vector_type(2))) float v2f;
typedef __attribute__((ext_vector_type(8))) float v8f;

#define NDIM  50
#define NN    2500     // NDIM*NDIM
#define ATT   4000
#define BB    512
#define PP    32
#define NPAIR (BB * PP)  // 16384

static __device__ __forceinline__ float wave_sum(float v) {
  #pragma unroll
  for (int m = 16; m > 0; m >>= 1) v += __shfl_xor(v, m, 32);
  return v;
}
static __device__ __forceinline__ float wave_max(float v) {
  #pragma unroll
  for (int m = 16; m > 0; m >>= 1) v = fmaxf(v, __shfl_xor(v, m, 32));
  return v;
}

// --------------------------- Kernel 1: F = g @ weights^2 -------------------
// One wave per output row; 2500 rows. Reads 25MB of g from HBM (the one
// unavoidable HBM stream), weights stays L2-hot.
__global__ void __launch_bounds__(256) k_compute_F(const float* __restrict__ g,
                                                   const float* __restrict__ w,
                                                   float* __restrict__ Fout) {
  const int lane = threadIdx.x & 31;
  const int row  = blockIdx.x * (blockDim.x >> 5) + (threadIdx.x >> 5);
  if (row >= NN) return;
  const float* grow = g + (size_t)row * NN;
  float s = 0.f;
  for (int j = lane; j < NN; j += 32) {
    const float wj = w[j];
    s = fmaf(grow[j], wj * wj, s);
  }
  s = wave_sum(s);
  if (lane == 0) Fout[row] = s;
}

// ------------------- Kernel 2: f = softmax(-alpha_t * F_row_i) -------------
// One wave per (t, i) row of 50; 200000 rows. Writes the 40MB f tensor
// (L2-resident for the gather phase).
__global__ void __launch_bounds__(256) k_softmax_f(const float* __restrict__ Fin,
                                                   const float* __restrict__ alphas,
                                                   float* __restrict__ f) {
  const int lane = threadIdx.x & 31;
  const int r    = blockIdx.x * (blockDim.x >> 5) + (threadIdx.x >> 5);
  if (r >= ATT * NDIM) return;
  const int t = r / NDIM;
  const int i = r % NDIM;
  const float alpha = alphas[t];
  const float* Frow = Fin + i * NDIM;
  const int j0 = lane, j1 = lane + 32;
  const float ninf = -__builtin_inff();
  float v0 = (j0 < NDIM) ? (-alpha * Frow[j0]) : ninf;
  float v1 = (j1 < NDIM) ? (-alpha * Frow[j1]) : ninf;
  const float m = wave_max(fmaxf(v0, v1));
  float e0 = (j0 < NDIM) ? __expf(v0 - m) : 0.f;
  float e1 = (j1 < NDIM) ? __expf(v1 - m) : 0.f;
  const float inv = 1.f / wave_sum(e0 + e1);
  float* frow = f + (size_t)t * NN + i * NDIM;
  if (j0 < NDIM) frow[j0] = e0 * inv;
  if (j1 < NDIM) frow[j1] = e1 * inv;
}

// ----------- Kernel 3: gathered per-pair matvec via WMMA f32 16x16x4 -------
// One wave per (b,p) pair (16384 waves, 8 waves/block, exact grid -> EXEC
// all-ones around every wmma). A = 16x4 tile of f[t] (M=i rows, K=j),
// B = a-vector broadcast across the 16 N-columns, D accumulated over 13
// K-steps of 4; column N=0 (lanes 0 and 16) carries z[b,p,i].
__global__ void __launch_bounds__(256) k_pair_matvec_wmma(
    const float* __restrict__ x, const int* __restrict__ x_i,
    const float* __restrict__ mus, const float* __restrict__ f,
    float* __restrict__ zpart) {
  __shared__ float sa[8][64];
  const int lane = threadIdx.x & 31;
  const int w    = threadIdx.x >> 5;
  const int q    = blockIdx.x * 8 + w;        // pair id in [0, NPAIR)
  const int b    = q >> 5;                    // q / PP
  const int p    = q & 31;                    // q % PP
  const int t    = x_i[q];                    // x_i[b][p]

  // a[j] = x[b, j, p] - mus[t, j], zero-padded to 64
  {
    const float* mrow = mus + (size_t)t * NDIM;
    const size_t xb = (size_t)b * NDIM * PP;
    const int j0 = lane, j1 = lane + 32;
    sa[w][j0] = (j0 < NDIM) ? (x[xb + (size_t)j0 * PP + p] - mrow[j0]) : 0.f;
    sa[w][j1] = (j1 < NDIM) ? (x[xb + (size_t)j1 * PP + p] - mrow[j1]) : 0.f;
  }
  const float* ft = f + (size_t)t * NN;
  // prefetch this pair's 10KB f-matrix (L2-resident) into near caches
  __builtin_prefetch(ft + lane * 80, 0, 0);   // 32 lanes x 320B covers 10KB
  __syncthreads();

  const int mrowi = lane & 15;                // M within 16-tile
  const int khalf = (lane >> 4) << 1;         // lanes 16-31 supply K+2,K+3

  #pragma unroll
  for (int mt = 0; mt < 4; ++mt) {
    const int i = mt * 16 + mrowi;
    const float* frow = ft + i * NDIM;
    v8f acc = {};
    #pragma unroll
    for (int kb = 0; kb < 52; kb += 4) {      // 13 K-steps, K padded 50->52
      const int j = kb + khalf;
      v2f av;
      if (i < NDIM && (j + 1) < NDIM) {
        av = *(const v2f*)(frow + j);         // 8B aligned: j even, row 200B
      } else {
        av = (v2f){0.f, 0.f};
      }
      v2f bv;
      bv.x = sa[w][j];                        // broadcast a across N columns
      bv.y = sa[w][j + 1];
      acc = __builtin_amdgcn_wmma_f32_16x16x4_f32(
          /*neg_a=*/false, av, /*neg_b=*/false, bv,
          /*c_mod=*/(short)0, acc, /*reuse_a=*/false, /*reuse_b=*/false);
    }
    // D layout: VGPR r -> M=r (lanes 0-15) / M=8+r (lanes 16-31), N=lane%16.
    // Column N==0 lives in lanes 0 and 16.
    if ((lane & 15) == 0) {
      const int ibase = mt * 16 + (lane >> 4) * 8;
      #pragma unroll
      for (int r = 0; r < 8; ++r) {
        const int ii = ibase + r;
        if (ii < NDIM) zpart[(size_t)q * NDIM + ii] = acc[r];
      }
    }
  }
}

// ---------------- Kernel 4: Z[b,i] = sum_p z[b,p,i] + P * mu_y[b,i] --------
__global__ void __launch_bounds__(256) k_reduce_Z(const float* __restrict__ zpart,
                                                  const int* __restrict__ y_i,
                                                  const float* __restrict__ mus,
                                                  float* __restrict__ Z) {
  const int o = blockIdx.x * blockDim.x + threadIdx.x;
  if (o >= BB * NDIM) return;
  const int b = o / NDIM;
  const int i = o % NDIM;
  const float* zp = zpart + (size_t)b * PP * NDIM + i;
  float s = 0.f;
  #pragma unroll 4
  for (int p = 0; p < PP; ++p) s += zp[(size_t)p * NDIM];   // deterministic order
  s += (float)PP * mus[(size_t)y_i[b] * NDIM + i];
  Z[o] = s;
}

// ---------------------------------------------------------------------------
extern "C" void kernel_launch(void* const* d_in, const int* in_sizes, int n_in,
                              void* d_out, int out_size, void* d_ws, size_t ws_size,
                              hipStream_t stream) {
  (void)in_sizes; (void)n_in; (void)out_size; (void)ws_size;
  const float* x       = (const float*)d_in[0];  // [512,50,32]
  const int*   x_i     = (const int*)  d_in[1];  // [512,32]
  const int*   y_i     = (const int*)  d_in[2];  // [512]
  const float* g       = (const float*)d_in[3];  // [2500,2500]
  const float* weights = (const float*)d_in[4];  // [2500,1]
  const float* mus     = (const float*)d_in[5];  // [4000,50]
  const float* alphas  = (const float*)d_in[6];  // [4000,1]

  float* Z    = (float*)d_out;                   // [512*50]
  float* Fout = (float*)d_out + BB * NDIM;       // [2500], second tuple output

  float* f     = (float*)d_ws;                   // [4000*2500] = 40MB (L2-hot)
  float* zpart = f + (size_t)ATT * NN;           // [16384*50]  = 3.3MB

  k_compute_F      <<<(NN + 7) / 8, 256, 0, stream>>>(g, weights, Fout);
  k_softmax_f      <<<(ATT * NDIM + 7) / 8, 256, 0, stream>>>(Fout, alphas, f);
  k_pair_matvec_wmma<<<NPAIR / 8, 256, 0, stream>>>(x, x_i, mus, f, zpart);
  k_reduce_Z       <<<(BB * NDIM + 255) / 256, 256, 0, stream>>>(zpart, y_i, mus, Z);
}